// MaIRUNet_28527172780364
// MI455X (gfx1250) — compile-verified
//
#include <hip/hip_runtime.h>

// ---------------------------------------------------------------------------
// MaIR / VMamba selective-scan block for MI455X (gfx1250, wave32).
// fp32 WMMA (V_WMMA_F32_16X16X4_F32) for all GEMMs; chunked 3-pass scan with
// double-buffered GLOBAL_LOAD_ASYNC_TO_LDS staging of the shared B/C tiles.
// ---------------------------------------------------------------------------

typedef __attribute__((ext_vector_type(2))) float v2f;
typedef __attribute__((ext_vector_type(8))) float v8f;

static constexpr int Hd = 64, Wd = 64, Cin = 64;
static constexpr int Di = 128, K = 4, R = 4, N = 16;
static constexpr int L  = Hd * Wd;      // 4096
static constexpr int C2 = 2 * Di;       // 256
static constexpr int CD = R + 2 * N;    // 36 rows of x_dbl
static constexpr int CDp = 48;          // padded to 3 WMMA tiles
static constexpr int SEG = 16, SEGLEN = L / SEG;   // 16 segments x 256
static constexpr float LN_EPS = 1e-5f;

// ---- CDNA5 async global->LDS path (guarded; falls back to sync staging) ----
#if defined(__has_builtin)
# if __has_builtin(__builtin_amdgcn_global_load_async_to_lds_b128) && \
     __has_builtin(__builtin_amdgcn_s_wait_asynccnt)
#  define USE_ASYNC_LDS 1
# endif
#endif
#ifndef USE_ASYNC_LDS
# define USE_ASYNC_LDS 0
#endif

#if USE_ASYNC_LDS
typedef int v4i_ __attribute__((vector_size(16)));   // matches builtin's V4i
#define GLOBAL_V4(p) ((__attribute__((address_space(1))) v4i_*)(p))
#define LDS_V4(p)    ((__attribute__((address_space(3))) v4i_*)(p))
#endif

__device__ __forceinline__ v8f wmma4(v2f a, v2f b, v8f c) {
  // D = A(16x4,f32) * B(4x16,f32) + C(16x16,f32)   [wave32]
  return __builtin_amdgcn_wmma_f32_16x16x4_f32(
      /*neg_a=*/false, a, /*neg_b=*/false, b,
      /*c_mod=*/(short)0, c, /*reuse_a=*/false, /*reuse_b=*/false);
}
__device__ __forceinline__ float sigmoidf_(float x) { return 1.f / (1.f + __expf(-x)); }
__device__ __forceinline__ float siluf_(float x)    { return x * sigmoidf_(x); }
__device__ __forceinline__ float softplusf_(float x){ return (x > 20.f) ? x : __logf(1.f + __expf(x)); }

// --------------------------- 1) in_proj GEMM --------------------------------
// xz[L,256] = x[L,64] @ W[256,64]^T ; split: cols<Di -> xin[d][l], else z[l][d]
__global__ __launch_bounds__(32)
void k_inproj(const float* __restrict__ x, const float* __restrict__ W,
              float* __restrict__ xin, float* __restrict__ z) {
  const int m0 = blockIdx.x * 16, n0 = blockIdx.y * 16;
  const int lane = threadIdx.x, half = lane >> 4, lr = lane & 15;
  const float* arow = x + (size_t)(m0 + lr) * Cin;   // A row = spatial l
  const float* brow = W + (size_t)(n0 + lr) * Cin;   // B "row" lane = out chan
  v8f acc = {};
  for (int kk = 0; kk < Cin; kk += 4) {
    v2f a, b;
    a.x = arow[kk + 2 * half];     a.y = arow[kk + 2 * half + 1];
    b.x = brow[kk + 2 * half];     b.y = brow[kk + 2 * half + 1];
    acc = wmma4(a, b, acc);
  }
#pragma unroll
  for (int j = 0; j < 8; ++j) {
    const int row = m0 + j + 8 * half;   // spatial l
    const int col = n0 + lr;             // channel
    const float v = acc[j];
    if (col < Di) xin[(size_t)col * L + row] = v;       // channel-major for conv
    else          z[(size_t)row * Di + (col - Di)] = v; // row-major for epilogue
  }
}

// --------------------- 2) depthwise 3x3 conv + SiLU -------------------------
__global__ void k_conv(const float* __restrict__ xin, const float* __restrict__ cw,
                       const float* __restrict__ cb, float* __restrict__ xc) {
  const int t = blockIdx.x * blockDim.x + threadIdx.x;
  if (t >= Di * L) return;
  const int d = t / L, hw = t - d * L, h = hw >> 6, w = hw & 63;
  const float* plane = xin + (size_t)d * L;
  const float* wt = cw + d * 9;
  float s = cb[d];
#pragma unroll
  for (int dy = -1; dy <= 1; ++dy) {
    const int hh = h + dy; if (hh < 0 || hh >= Hd) continue;
#pragma unroll
    for (int dx = -1; dx <= 1; ++dx) {
      const int ww = w + dx; if (ww < 0 || ww >= Wd) continue;
      s += plane[hh * Wd + ww] * wt[(dy + 1) * 3 + (dx + 1)];
    }
  }
  xc[t] = siluf_(s);
}

// ------------------------- 3) scan-order gather -----------------------------
__global__ void k_gather(const float* __restrict__ xc, const int* __restrict__ ids,
                         float* __restrict__ xs) {
  const int t = blockIdx.x * blockDim.x + threadIdx.x;
  if (t >= K * Di * L) return;
  const int l = t % L, kd = t / L, d = kd % Di, k = kd / Di;
  xs[t] = xc[(size_t)d * L + ids[k * L + l]];
}

// ---------------------- 4) x_dbl = W_k[36,Di] @ xs_k[Di,L] ------------------
__global__ __launch_bounds__(32)
void k_xdbl(const float* __restrict__ xpw, const float* __restrict__ xs,
            float* __restrict__ xdbl) {
  const int m0 = blockIdx.x * 16, n0 = blockIdx.y * 16, k = blockIdx.z;
  const int lane = threadIdx.x, half = lane >> 4, lr = lane & 15;
  const float* Wk = xpw + (size_t)k * CD * Di;
  const float* Xk = xs + (size_t)k * Di * L;
  const bool arow_ok = (m0 + lr) < CD;     // zero-pad rows 36..47
  const float* arow = Wk + (size_t)(m0 + lr) * Di;
  v8f acc = {};
  for (int kk = 0; kk < Di; kk += 4) {
    v2f a, b;
    a.x = arow_ok ? arow[kk + 2 * half]     : 0.f;
    a.y = arow_ok ? arow[kk + 2 * half + 1] : 0.f;
    b.x = Xk[(size_t)(kk + 2 * half) * L + n0 + lr];
    b.y = Xk[(size_t)(kk + 2 * half + 1) * L + n0 + lr];
    acc = wmma4(a, b, acc);
  }
#pragma unroll
  for (int j = 0; j < 8; ++j) {
    const int row = m0 + j + 8 * half;
    if (row < CD) xdbl[((size_t)k * CD + row) * L + n0 + lr] = acc[j];
  }
}

// -------- 5) delta = softplus(dtw_k[Di,4] @ dts_k[4,L] + bias) (1 WMMA) -----
__global__ __launch_bounds__(32)
void k_delta(const float* __restrict__ dtw, const float* __restrict__ xdbl,
             const float* __restrict__ dtb, float* __restrict__ delta) {
  const int m0 = blockIdx.x * 16, n0 = blockIdx.y * 16, k = blockIdx.z;
  const int lane = threadIdx.x, half = lane >> 4, lr = lane & 15;
  const float* Wk = dtw + (size_t)k * Di * R;
  v2f a, b;
  a.x = Wk[(m0 + lr) * R + 2 * half];
  a.y = Wk[(m0 + lr) * R + 2 * half + 1];
  b.x = xdbl[((size_t)k * CD + 2 * half)     * L + n0 + lr];
  b.y = xdbl[((size_t)k * CD + 2 * half + 1) * L + n0 + lr];
  v8f acc = {};
  acc = wmma4(a, b, acc);
#pragma unroll
  for (int j = 0; j < 8; ++j) {
    const int d = m0 + j + 8 * half;
    delta[((size_t)k * Di + d) * L + n0 + lr] = softplusf_(acc[j] + dtb[k * Di + d]);
  }
}

// -------------------------- A = -exp(A_logs) --------------------------------
__global__ void k_prep(const float* __restrict__ Alogs, float* __restrict__ Amat) {
  const int t = blockIdx.x * blockDim.x + threadIdx.x;
  if (t < K * Di * N) Amat[t] = -__expf(Alogs[t]);
}

// -------- 6) scan pass 1: per-segment (sum_delta, B-accumulator) ------------
__global__ __launch_bounds__(128)
void k_scan1(const float* __restrict__ delta, const float* __restrict__ xs,
             const float* __restrict__ xdbl, const float* __restrict__ Amat,
             float* __restrict__ segSum, float* __restrict__ segB) {
  __shared__ float tile[2][N * 64];          // [buf][n*64 + j]
  const int s = blockIdx.x, k = blockIdx.y, d = threadIdx.x;
  const size_t kd = (size_t)k * Di + d;
  const float* dptr = delta + kd * L;
  const float* xptr = xs + kd * L;
  const float* gB = xdbl + ((size_t)k * CD + R) * L;   // Bs row n at gB + n*L
  float Areg[N], hb[N];
#pragma unroll
  for (int n = 0; n < N; ++n) { Areg[n] = Amat[kd * N + n]; hb[n] = 0.f; }
  float sumd = 0.f;
  const int l0 = s * SEGLEN;
  constexpr int NCH = SEGLEN / 64;           // 4 chunks of 64 steps

#if USE_ASYNC_LDS
  auto stage = [&](int buf, int lc) {
#pragma unroll
    for (int i = 0; i < 2; ++i) {            // 2 wave-wide B128 issues / chunk
      const int e = i * 128 + (int)threadIdx.x;     // 0..255, 4 floats each
      const int n = e >> 4, j4 = (e & 15) << 2;
      __builtin_amdgcn_global_load_async_to_lds_b128(
          GLOBAL_V4(gB + (size_t)n * L + lc + j4),
          LDS_V4(&tile[buf][e << 2]),
          0, 0);
    }
  };
  stage(0, l0);
  for (int c = 0; c < NCH; ++c) {
    const int lc = l0 + c * 64;
    if (c + 1 < NCH) { stage((c + 1) & 1, lc + 64);
                       __builtin_amdgcn_s_wait_asynccnt(2); }
    else             { __builtin_amdgcn_s_wait_asynccnt(0); }
    __syncthreads();
    const float* sB = tile[c & 1];
    for (int j = 0; j < 64; ++j) {
      const float dl = dptr[lc + j];
      const float du = dl * xptr[lc + j];
      sumd += dl;
#pragma unroll
      for (int n = 0; n < N; ++n)
        hb[n] = __expf(dl * Areg[n]) * hb[n] + du * sB[n * 64 + j];
    }
    __syncthreads();
  }
#else
  for (int c = 0; c < NCH; ++c) {
    const int lc = l0 + c * 64;
    __syncthreads();
    for (int i = threadIdx.x; i < N * 64; i += 128) {
      const int n = i >> 6, j = i & 63;
      tile[0][n * 64 + j] = gB[(size_t)n * L + lc + j];
    }
    __syncthreads();
    const float* sB = tile[0];
    for (int j = 0; j < 64; ++j) {
      const float dl = dptr[lc + j];
      const float du = dl * xptr[lc + j];
      sumd += dl;
#pragma unroll
      for (int n = 0; n < N; ++n)
        hb[n] = __expf(dl * Areg[n]) * hb[n] + du * sB[n * 64 + j];
    }
  }
#endif
  segSum[kd * SEG + s] = sumd;
  float* sb = segB + (kd * SEG + s) * N;
#pragma unroll
  for (int n = 0; n < N; ++n) sb[n] = hb[n];
}

// -------- 7) scan pass 2: compose 16 segment states per channel -------------
__global__ __launch_bounds__(128)
void k_scan2(const float* __restrict__ segSum, const float* __restrict__ segB,
             const float* __restrict__ Amat, float* __restrict__ hinit) {
  const int k = blockIdx.x, d = threadIdx.x;
  const size_t kd = (size_t)k * Di + d;
  float Areg[N], h[N];
#pragma unroll
  for (int n = 0; n < N; ++n) { Areg[n] = Amat[kd * N + n]; h[n] = 0.f; }
  for (int s = 0; s < SEG; ++s) {
    float* hp = hinit + (kd * SEG + s) * N;
#pragma unroll
    for (int n = 0; n < N; ++n) hp[n] = h[n];
    const float sd = segSum[kd * SEG + s];
    const float* sb = segB + (kd * SEG + s) * N;
#pragma unroll
    for (int n = 0; n < N; ++n) h[n] = __expf(Areg[n] * sd) * h[n] + sb[n];
  }
}

// -------- 8) scan pass 3: replay with true h0, emit y -----------------------
__global__ __launch_bounds__(128)
void k_scan3(const float* __restrict__ delta, const float* __restrict__ xs,
             const float* __restrict__ xdbl, const float* __restrict__ Amat,
             const float* __restrict__ hinit, const float* __restrict__ Dsv,
             float* __restrict__ yscan) {
  __shared__ float tile[2][2 * N * 64];      // [buf][ (B:0..15 | C:16..31)*64 + j ]
  const int s = blockIdx.x, k = blockIdx.y, d = threadIdx.x;
  const size_t kd = (size_t)k * Di + d;
  const float* dptr = delta + kd * L;
  const float* xptr = xs + kd * L;
  const float* gBC = xdbl + ((size_t)k * CD + R) * L;  // rows: 16 Bs then 16 Cs
  const float* hp = hinit + (kd * SEG + s) * N;
  float Areg[N], h[N];
#pragma unroll
  for (int n = 0; n < N; ++n) { Areg[n] = Amat[kd * N + n]; h[n] = hp[n]; }
  const float dskd = Dsv[kd];
  const int l0 = s * SEGLEN;
  constexpr int NCH = SEGLEN / 64;

#if USE_ASYNC_LDS
  auto stage = [&](int buf, int lc) {
#pragma unroll
    for (int i = 0; i < 4; ++i) {            // 4 wave-wide B128 issues / chunk
      const int e = i * 128 + (int)threadIdx.x;     // 0..511, 4 floats each
      const int n = e >> 4, j4 = (e & 15) << 2;     // n: 0..31 (B then C rows)
      __builtin_amdgcn_global_load_async_to_lds_b128(
          GLOBAL_V4(gBC + (size_t)n * L + lc + j4),
          LDS_V4(&tile[buf][e << 2]),
          0, 0);
    }
  };
  stage(0, l0);
  for (int c = 0; c < NCH; ++c) {
    const int lc = l0 + c * 64;
    if (c + 1 < NCH) { stage((c + 1) & 1, lc + 64);
                       __builtin_amdgcn_s_wait_asynccnt(4); }
    else             { __builtin_amdgcn_s_wait_asynccnt(0); }
    __syncthreads();
    const float* tb = tile[c & 1];
    for (int j = 0; j < 64; ++j) {
      const float dl = dptr[lc + j];
      const float xv = xptr[lc + j];
      const float du = dl * xv;
      float y = 0.f;
#pragma unroll
      for (int n = 0; n < N; ++n) {
        h[n] = __expf(dl * Areg[n]) * h[n] + du * tb[n * 64 + j];
        y += h[n] * tb[(N + n) * 64 + j];
      }
      yscan[kd * L + lc + j] = y + dskd * xv;
    }
    __syncthreads();
  }
#else
  for (int c = 0; c < NCH; ++c) {
    const int lc = l0 + c * 64;
    __syncthreads();
    for (int i = threadIdx.x; i < 2 * N * 64; i += 128) {
      const int n = i >> 6, j = i & 63;
      tile[0][n * 64 + j] = gBC[(size_t)n * L + lc + j];
    }
    __syncthreads();
    const float* tb = tile[0];
    for (int j = 0; j < 64; ++j) {
      const float dl = dptr[lc + j];
      const float xv = xptr[lc + j];
      const float du = dl * xv;
      float y = 0.f;
#pragma unroll
      for (int n = 0; n < N; ++n) {
        h[n] = __expf(dl * Areg[n]) * h[n] + du * tb[n * 64 + j];
        y += h[n] * tb[(N + n) * 64 + j];
      }
      yscan[kd * L + lc + j] = y + dskd * xv;
    }
  }
#endif
}

// ------------- 9) channel means (deterministic block reduction) -------------
__global__ __launch_bounds__(256)
void k_pool(const float* __restrict__ yscan, float* __restrict__ pooled) {
  __shared__ float red[256];
  const int ch = blockIdx.x;                       // k*Di + d
  const float* p = yscan + (size_t)ch * L;
  float sacc = 0.f;
  for (int l = threadIdx.x; l < L; l += 256) sacc += p[l];
  red[threadIdx.x] = sacc;
  __syncthreads();
  for (int off = 128; off; off >>= 1) {
    if (threadIdx.x < off) red[threadIdx.x] += red[threadIdx.x + off];
    __syncthreads();
  }
  if (threadIdx.x == 0) pooled[ch] = red[0] * (1.f / L);
}

// --------------------------- 10) gating -------------------------------------
// gate(k,d) = sigmoid( sum_i gw[d*4+k][i] * pooled[i*Di+d] + gb[d*4+k] )
__global__ void k_gate(const float* __restrict__ pooled, const float* __restrict__ gw,
                       const float* __restrict__ gb, float* __restrict__ gate) {
  const int t = blockIdx.x * blockDim.x + threadIdx.x;
  if (t >= K * Di) return;
  const int k = t / Di, d = t - k * Di;
  float sacc = gb[d * 4 + k];
#pragma unroll
  for (int i = 0; i < 4; ++i)
    sacc += gw[(d * 4 + k) * 4 + i] * pooled[i * Di + d];
  gate[t] = sigmoidf_(sacc);
}

// ----- 11) inverse-permute + gated K-sum + LayerNorm + *silu(z) -------------
__global__ __launch_bounds__(128)
void k_combine(const float* __restrict__ yscan, const int* __restrict__ inv,
               const float* __restrict__ gate, const float* __restrict__ z,
               const float* __restrict__ nw, const float* __restrict__ nb,
               float* __restrict__ gbuf) {
  __shared__ float red[128];
  const int j = blockIdx.x, d = threadIdx.x;
  float v = 0.f;
#pragma unroll
  for (int k = 0; k < K; ++k) {
    const int l = inv[k * L + j];
    v += gate[k * Di + d] * yscan[((size_t)k * Di + d) * L + l];
  }
  red[d] = v; __syncthreads();
  for (int off = 64; off; off >>= 1) {
    if (d < off) red[d] += red[d + off];
    __syncthreads();
  }
  const float mu = red[0] * (1.f / Di);
  __syncthreads();
  const float c = v - mu;
  red[d] = c * c; __syncthreads();
  for (int off = 64; off; off >>= 1) {
    if (d < off) red[d] += red[d + off];
    __syncthreads();
  }
  const float var = red[0] * (1.f / Di);
  const float yn = c * rsqrtf(var + LN_EPS) * nw[d] + nb[d];
  gbuf[(size_t)j * Di + d] = yn * siluf_(z[(size_t)j * Di + d]);
}

// --------------------- 12) out_proj GEMM [L,Di]x[Di,64] ---------------------
__global__ __launch_bounds__(32)
void k_outproj(const float* __restrict__ g, const float* __restrict__ Wout,
               float* __restrict__ out) {
  const int m0 = blockIdx.x * 16, n0 = blockIdx.y * 16;
  const int lane = threadIdx.x, half = lane >> 4, lr = lane & 15;
  const float* arow = g + (size_t)(m0 + lr) * Di;
  const float* brow = Wout + (size_t)(n0 + lr) * Di;
  v8f acc = {};
  for (int kk = 0; kk < Di; kk += 4) {
    v2f a, b;
    a.x = arow[kk + 2 * half];  a.y = arow[kk + 2 * half + 1];
    b.x = brow[kk + 2 * half];  b.y = brow[kk + 2 * half + 1];
    acc = wmma4(a, b, acc);
  }
#pragma unroll
  for (int j = 0; j < 8; ++j)
    out[(size_t)(m0 + j + 8 * half) * Cin + n0 + lr] = acc[j];
}

// ---------------------------------------------------------------------------
extern "C" void kernel_launch(void* const* d_in, const int* in_sizes, int n_in,
                              void* d_out, int out_size, void* d_ws, size_t ws_size,
                              hipStream_t stream) {
  (void)in_sizes; (void)n_in; (void)out_size; (void)ws_size;
  const float* x     = (const float*)d_in[0];
  const float* inpw  = (const float*)d_in[1];
  const float* convw = (const float*)d_in[2];
  const float* convb = (const float*)d_in[3];
  const float* xpw   = (const float*)d_in[4];
  const float* dtw   = (const float*)d_in[5];
  const float* dtb   = (const float*)d_in[6];
  const float* alogs = (const float*)d_in[7];
  const float* dsv   = (const float*)d_in[8];
  const float* gw    = (const float*)d_in[9];
  const float* gb    = (const float*)d_in[10];
  const float* nw    = (const float*)d_in[11];
  const float* nb    = (const float*)d_in[12];
  const float* outw  = (const float*)d_in[13];
  const int*   sids  = (const int*)d_in[14];
  const int*   iids  = (const int*)d_in[15];
  float* out = (float*)d_out;

  float* ws = (float*)d_ws;
  size_t off = 0;
  auto alloc = [&](size_t nf) { float* p = ws + off; off += nf; return p; };
  float* xin   = alloc((size_t)Di * L);
  float* z     = alloc((size_t)L * Di);
  float* xc    = alloc((size_t)Di * L);
  float* xs    = alloc((size_t)K * Di * L);
  float* xdbl  = alloc((size_t)K * CD * L);
  float* delta = alloc((size_t)K * Di * L);
  float* Amat  = alloc((size_t)K * Di * N);
  float* segS  = alloc((size_t)K * Di * SEG);
  float* segB  = alloc((size_t)K * Di * SEG * N);
  float* hini  = alloc((size_t)K * Di * SEG * N);
  float* ysc   = alloc((size_t)K * Di * L);
  float* pooled= alloc((size_t)K * Di);
  float* gate  = alloc((size_t)K * Di);
  float* gbuf  = alloc((size_t)L * Di);

  k_inproj <<<dim3(L / 16, C2 / 16), 32, 0, stream>>>(x, inpw, xin, z);
  k_conv   <<<(Di * L + 255) / 256, 256, 0, stream>>>(xin, convw, convb, xc);
  k_gather <<<(K * Di * L + 255) / 256, 256, 0, stream>>>(xc, sids, xs);
  k_xdbl   <<<dim3(CDp / 16, L / 16, K), 32, 0, stream>>>(xpw, xs, xdbl);
  k_delta  <<<dim3(Di / 16, L / 16, K), 32, 0, stream>>>(dtw, xdbl, dtb, delta);
  k_prep   <<<(K * Di * N + 255) / 256, 256, 0, stream>>>(alogs, Amat);
  k_scan1  <<<dim3(SEG, K), 128, 0, stream>>>(delta, xs, xdbl, Amat, segS, segB);
  k_scan2  <<<K, 128, 0, stream>>>(segS, segB, Amat, hini);
  k_scan3  <<<dim3(SEG, K), 128, 0, stream>>>(delta, xs, xdbl, Amat, hini, dsv, ysc);
  k_pool   <<<K * Di, 256, 0, stream>>>(ysc, pooled);
  k_gate   <<<(K * Di + 255) / 256, 256, 0, stream>>>(pooled, gw, gb, gate);
  k_combine<<<L, 128, 0, stream>>>(ysc, iids, gate, z, nw, nb, gbuf);
  k_outproj<<<dim3(L / 16, Cin / 16), 32, 0, stream>>>(gbuf, outw, out);
}